// GCNLayer_70489003262548
// MI455X (gfx1250) — compile-verified
//
#include <hip/hip_runtime.h>
#include <hip/hip_bf16.h>

typedef __attribute__((ext_vector_type(2))) float v2f;
typedef __attribute__((ext_vector_type(8))) float v8f;

#define CH 128          // IN_CH == OUT_CH == 128
#define BN_EPS 1e-5f

// ---------------- degree / normalization ----------------

__global__ void k_deg_init(float* __restrict__ deg, int n) {
    int i = blockIdx.x * blockDim.x + threadIdx.x;
    if (i < n) deg[i] = 1.0f;   // self-loop contributes 1
}

__global__ void k_deg_count(const int* __restrict__ dst, float* __restrict__ deg, int ne) {
    int e = blockIdx.x * blockDim.x + threadIdx.x;
    if (e < ne) atomicAdd(&deg[dst[e]], 1.0f);
}

__global__ void k_dinv(const float* __restrict__ deg, float* __restrict__ dinv, int n) {
    int i = blockIdx.x * blockDim.x + threadIdx.x;
    if (i < n) dinv[i] = rsqrtf(deg[i]);   // deg >= 1 always (self-loops)
}

// ---------------- GEMM: xw = x @ W via V_WMMA_F32_16X16X4_F32 ----------------
// 256 threads = 8 waves per block; wave w owns rows [strip*16, strip*16+16),
// computes all 128 output columns as 8 16x16 f32 accumulator tiles.
// A 16x4 f32 layout: lane = M + 16*khalf, VGPR j holds K = 2*khalf + j.
// B 4x16  f32 layout (mirror): lane = N + 16*khalf, VGPR j holds K = 2*khalf + j.
// C/D 16x16 f32: VGPR r -> row (r + 8*khalf), col = lane & 15.

__global__ void k_gemm_wmma(const float* __restrict__ x, const float* __restrict__ W,
                            float* __restrict__ xw, int n_strips) {
    const int wave = threadIdx.x >> 5;
    const int lane = threadIdx.x & 31;
    const int strip = blockIdx.x * 8 + wave;
    if (strip >= n_strips) return;              // wave-uniform: EXEC stays all-1s
    const int row0 = strip * 16;
    const int m = lane & 15;
    const int khalf = lane >> 4;                // 0 or 1

    v8f zero = {};
    v8f acc[8];
#pragma unroll
    for (int t = 0; t < 8; ++t) acc[t] = zero;

    const float* __restrict__ arow = x + (size_t)(row0 + m) * CH;

    for (int kb = 0; kb < CH; kb += 4) {
        const int ka = kb + 2 * khalf;
        v2f a;
        a.x = arow[ka + 0];
        a.y = arow[ka + 1];
#pragma unroll
        for (int t = 0; t < 8; ++t) {
            const int n0 = t * 16;
            v2f bf;
            bf.x = W[(size_t)(ka + 0) * CH + n0 + m];
            bf.y = W[(size_t)(ka + 1) * CH + n0 + m];
            acc[t] = __builtin_amdgcn_wmma_f32_16x16x4_f32(
                false, a, false, bf, (short)0, acc[t], false, false);
        }
    }

#pragma unroll
    for (int t = 0; t < 8; ++t) {
        const int col = t * 16 + m;
#pragma unroll
        for (int r = 0; r < 8; ++r) {
            xw[(size_t)(row0 + r + 8 * khalf) * CH + col] = acc[t][r];
        }
    }
}

// scalar fallback for a ragged tail (n % 16 != 0) — not hit for N=100000
__global__ void k_gemm_rem(const float* __restrict__ x, const float* __restrict__ W,
                           float* __restrict__ xw, int row_start, int n_rows) {
    int idx = blockIdx.x * blockDim.x + threadIdx.x;
    int total = (n_rows - row_start) * CH;
    if (idx >= total) return;
    int r = row_start + idx / CH;
    int c = idx % CH;
    float acc = 0.0f;
    for (int k = 0; k < CH; ++k) acc += x[(size_t)r * CH + k] * W[(size_t)k * CH + c];
    xw[(size_t)r * CH + c] = acc;
}

// ---------------- self-loop init + stats zero ----------------

__global__ void k_init_out(const float* __restrict__ xw, const float* __restrict__ dinv,
                           const float* __restrict__ b, float* __restrict__ out, int n) {
    int idx = blockIdx.x * blockDim.x + threadIdx.x;
    if (idx >= n * CH) return;
    int r = idx >> 7;
    int c = idx & (CH - 1);
    float di = dinv[r];
    out[idx] = xw[idx] * (di * di) + b[c];
}

__global__ void k_zero_stats(float* __restrict__ stats) {
    int i = threadIdx.x;        // 256 threads: sums[128] + sumsq[128]
    stats[i] = 0.0f;
}

// ---------------- edge scatter: 1 wave per edge, 4 channels per lane ----------------

__global__ void k_scatter(const int* __restrict__ src, const int* __restrict__ dst,
                          const float* __restrict__ dinv, const float* __restrict__ xw,
                          float* __restrict__ out, int ne) {
    int e = blockIdx.x * (blockDim.x >> 5) + (threadIdx.x >> 5);
    if (e >= ne) return;
    int lane = threadIdx.x & 31;
    int s = src[e];
    int d = dst[e];
    float nrm = dinv[s] * dinv[d];
    const float4* __restrict__ g = (const float4*)(xw + (size_t)s * CH);
    float4 v = g[lane];                          // coalesced 512B per wave
    float* __restrict__ o = out + (size_t)d * CH + lane * 4;
    atomicAdd(o + 0, v.x * nrm);
    atomicAdd(o + 1, v.y * nrm);
    atomicAdd(o + 2, v.z * nrm);
    atomicAdd(o + 3, v.w * nrm);
}

// ---------------- batch-norm statistics + apply ----------------

__global__ void k_stats(const float* __restrict__ out, float* __restrict__ sums,
                        float* __restrict__ sumsq, int n) {
    int c = threadIdx.x;                         // 128 threads = 1 channel each
    float s = 0.0f, s2 = 0.0f;
    for (int r = blockIdx.x; r < n; r += gridDim.x) {
        float v = out[(size_t)r * CH + c];
        s += v;
        s2 += v * v;
    }
    atomicAdd(&sums[c], s);
    atomicAdd(&sumsq[c], s2);
}

__global__ void k_bn_relu(float* __restrict__ out, const float* __restrict__ sums,
                          const float* __restrict__ sumsq, const float* __restrict__ gamma,
                          const float* __restrict__ beta, int n) {
    int idx = blockIdx.x * blockDim.x + threadIdx.x;
    if (idx >= n * CH) return;
    int c = idx & (CH - 1);
    float inv_n = 1.0f / (float)n;
    float mean = sums[c] * inv_n;
    float var = sumsq[c] * inv_n - mean * mean;  // biased variance (matches jnp.var)
    float v = (out[idx] - mean) * rsqrtf(var + BN_EPS) * gamma[c] + beta[c];
    out[idx] = fmaxf(v, 0.0f);
}

// ---------------- launch ----------------

extern "C" void kernel_launch(void* const* d_in, const int* in_sizes, int n_in,
                              void* d_out, int out_size, void* d_ws, size_t ws_size,
                              hipStream_t stream) {
    const float* x     = (const float*)d_in[0];
    const int*   eidx  = (const int*)d_in[1];
    const float* W     = (const float*)d_in[2];
    const float* b     = (const float*)d_in[3];
    const float* gamma = (const float*)d_in[4];
    const float* beta  = (const float*)d_in[5];

    const int n  = in_sizes[0] / CH;   // 100000 nodes
    const int ne = in_sizes[1] / 2;    // 1600000 edges
    const int* src = eidx;             // edge_index[0]
    const int* dst = eidx + ne;        // edge_index[1]

    float* out = (float*)d_out;        // [n, CH]; pre-BN accumulation done in place

    // workspace layout
    float* xw    = (float*)d_ws;                 // n*CH
    float* deg   = xw + (size_t)n * CH;          // n
    float* dinv  = deg + n;                      // n
    float* sums  = dinv + n;                     // CH
    float* sumsq = sums + CH;                    // CH

    const int T = 256;

    // 1) degrees (dst side, incl. self-loop) -> rsqrt
    k_deg_init<<<(n + T - 1) / T, T, 0, stream>>>(deg, n);
    k_deg_count<<<(ne + T - 1) / T, T, 0, stream>>>(dst, deg, ne);
    k_dinv<<<(n + T - 1) / T, T, 0, stream>>>(deg, dinv, n);

    // 2) xw = x @ W  (WMMA f32 16x16x4)
    const int n_strips = n / 16;
    if (n_strips > 0)
        k_gemm_wmma<<<(n_strips + 7) / 8, T, 0, stream>>>(x, W, xw, n_strips);
    const int rem_start = n_strips * 16;
    if (rem_start < n) {
        int rem_elems = (n - rem_start) * CH;
        k_gemm_rem<<<(rem_elems + T - 1) / T, T, 0, stream>>>(x, W, xw, rem_start, n);
    }

    // 3) out = xw * dinv^2 + b (self-loop term + bias); zero stat accumulators
    k_init_out<<<((size_t)n * CH + T - 1) / T, T, 0, stream>>>(xw, dinv, b, out, n);
    k_zero_stats<<<1, 256, 0, stream>>>(sums);

    // 4) edge scatter-add (atomics, mostly L2-resident)
    k_scatter<<<(ne + 7) / 8, T, 0, stream>>>(src, dst, dinv, xw, out, ne);

    // 5) batch-norm stats + apply + ReLU (in place on d_out)
    k_stats<<<512, CH, 0, stream>>>(out, sums, sumsq, n);
    k_bn_relu<<<((size_t)n * CH + T - 1) / T, T, 0, stream>>>(out, sums, sumsq, gamma, beta, n);
}